// Demosaic_4037269258504
// MI455X (gfx1250) — compile-verified
//
#include <hip/hip_runtime.h>
#include <hip/hip_bf16.h>

// ---------------------------------------------------------------------------
// Malvar-He-Cutler demosaic, MI455X (gfx1250).
// Bandwidth-bound (~268 MB moved, ~11.5 us floor @ 23.3 TB/s), so the
// optimization is the data path: async global->LDS staging (ASYNCcnt) +
// LDS-resident 5x5 stencil + non-temporal output stores.
// ---------------------------------------------------------------------------

#define IMG_H 1024
#define IMG_W 1024
#define IMG_B 16

#define TX 64              // output tile width
#define TY 16              // output tile height
#define LW (TX + 4)        // 68 floats per LDS row (272 B, 16B aligned)
#define LH (TY + 4)        // 20 LDS rows
#define NTHREADS 256

typedef __attribute__((address_space(1))) int g_int;
typedef __attribute__((address_space(3))) int l_int;

// ---- gfx1250 async global->LDS copy (one f32 per active lane) -------------
__device__ __forceinline__ void async_copy_f32(const float* src, float* dst_lds) {
#if defined(__AMDGCN__)
#if __has_builtin(__builtin_amdgcn_global_load_async_to_lds_b32)
    __builtin_amdgcn_global_load_async_to_lds_b32(
        (g_int*)src,
        (l_int*)dst_lds,
        /*offset=*/0, /*cpol=*/0);
#else
    // GV mode: 64-bit global address in VGPR pair, LDS byte address in VGPR.
    unsigned lds_off = (unsigned)(unsigned long long)dst_lds; // low 32 bits = LDS offset
    unsigned long long gaddr = (unsigned long long)src;
    asm volatile("global_load_async_to_lds_b32 %0, %1, off"
                 :
                 : "v"(lds_off), "v"(gaddr)
                 : "memory");
#endif
#else
    (void)src; (void)dst_lds;
#endif
}

__device__ __forceinline__ void wait_async0() {
#if defined(__AMDGCN__)
#if __has_builtin(__builtin_amdgcn_s_wait_asynccnt)
    __builtin_amdgcn_s_wait_asynccnt(0);
#else
    asm volatile("s_wait_asynccnt 0" ::: "memory");
#endif
#endif
}

__global__ __launch_bounds__(NTHREADS)
void demosaic_kernel(const float* __restrict__ cfa,
                     const float* __restrict__ k1,   // GR_GB
                     const float* __restrict__ k2,   // Rg_RB_Bg_BR
                     const float* __restrict__ k3,   // Rg_BR_Bg_RB
                     const float* __restrict__ k4,   // Rb_BB_Br_RR
                     float* __restrict__ out)
{
    __shared__ float tile[LH * LW];

    const int x0 = blockIdx.x * TX;
    const int y0 = blockIdx.y * TY;
    const int b  = blockIdx.z;

    const float* img = cfa + (size_t)b * (IMG_H * IMG_W);
    const int tid = threadIdx.x;

    // ---- stage input tile (2-halo, zero padded at borders) into LDS -------
    for (int i = tid; i < LH * LW; i += NTHREADS) {
        int ly = i / LW;
        int lx = i - ly * LW;
        int gy = y0 + ly - 2;
        int gx = x0 + lx - 2;
        if (gy >= 0 && gy < IMG_H && gx >= 0 && gx < IMG_W) {
            async_copy_f32(&img[(size_t)gy * IMG_W + gx], &tile[i]);
        } else {
            tile[i] = 0.0f;   // matches conv zero padding
        }
    }
    wait_async0();
    __syncthreads();

    // ---- weights (uniform scalar loads; kernels already include the /8) ---
    // GR_GB: center, +-1 cross, +-2 cross
    const float gC = k1[12], g1 = k1[7],  g2 = k1[2];
    // Rg_RB_Bg_BR: center, +-1 horiz, +-2 horiz, +-2 vert, diagonals
    const float hC = k2[12], h1 = k2[11], hE = k2[10], hN = k2[2], hD = k2[6];
    // Rg_BR_Bg_RB (transpose): center, +-1 vert, +-2 vert, +-2 horiz, diag
    const float vC = k3[12], v1 = k3[7],  vN = k3[2],  vE = k3[10], vD = k3[6];
    // Rb_BB_Br_RR: center, diagonals, +-2 cross
    const float xC = k4[12], xD = k4[6],  xN = k4[2],  xE = k4[10];

    float* __restrict__ outR = out + ((size_t)b * 3 + 0) * (IMG_H * IMG_W);
    float* __restrict__ outG = out + ((size_t)b * 3 + 1) * (IMG_H * IMG_W);
    float* __restrict__ outB = out + ((size_t)b * 3 + 2) * (IMG_H * IMG_W);

    // thread -> one column, 4 consecutive rows (vertical reuse of LDS rows)
    const int tx  = tid & (TX - 1);        // 0..63
    const int ty0 = (tid >> 6) * 4;        // 0,4,8,12
    const int ox  = x0 + tx;
    const int oddx = ox & 1;

#pragma unroll
    for (int k = 0; k < 4; ++k) {
        const int oy = y0 + ty0 + k;
        const float* t = &tile[(size_t)(ty0 + k + 2) * LW + (tx + 2)];

        const float c  = t[0];
        const float w1 = t[-1],      e1 = t[1];
        const float w2 = t[-2],      e2 = t[2];
        const float n1 = t[-LW],     s1 = t[LW];
        const float n2 = t[-2*LW],   s2 = t[2*LW];
        const float nw = t[-LW - 1], ne = t[-LW + 1];
        const float sw = t[ LW - 1], se = t[ LW + 1];

        const float hs1 = w1 + e1;            // horizontal +-1 sum
        const float hs2 = w2 + e2;            // horizontal +-2 sum
        const float vs1 = n1 + s1;            // vertical   +-1 sum
        const float vs2 = n2 + s2;            // vertical   +-2 sum
        const float dg  = (nw + ne) + (sw + se);

        // the four Malvar interpolants
        const float convG = gC * c + g1 * (hs1 + vs1) + g2 * (hs2 + vs2);
        const float convH = hC * c + h1 * hs1 + hE * hs2 + hN * vs2 + hD * dg;
        const float convV = vC * c + v1 * vs1 + vN * vs2 + vE * hs2 + vD * dg;
        const float convX = xC * c + xD * dg  + xN * vs2 + xE * hs2;

        const int oddy = oy & 1;
        // RGGB selection, branchless:
        //  (e,e) R-site:  R=c      G=convG  B=convX
        //  (e,o) Gr-site: R=convH  G=c      B=convV
        //  (o,e) Gb-site: R=convV  G=c      B=convH
        //  (o,o) B-site:  R=convX  G=convG  B=c
        const float Rv = oddy ? (oddx ? convX : convV) : (oddx ? convH : c);
        const float Gv = (oddx ^ oddy) ? c : convG;
        const float Bv = oddy ? (oddx ? c : convH) : (oddx ? convV : convX);

        const size_t o = (size_t)oy * IMG_W + ox;
        __builtin_nontemporal_store(Rv, &outR[o]);  // write-once planes:
        __builtin_nontemporal_store(Gv, &outG[o]);  // NT keeps 192 MB of
        __builtin_nontemporal_store(Bv, &outB[o]);  // output out of L2's way
    }
}

extern "C" void kernel_launch(void* const* d_in, const int* in_sizes, int n_in,
                              void* d_out, int out_size, void* d_ws, size_t ws_size,
                              hipStream_t stream) {
    (void)in_sizes; (void)n_in; (void)out_size; (void)d_ws; (void)ws_size;

    const float* cfa = (const float*)d_in[0];
    const float* k1  = (const float*)d_in[1];   // GR_GB
    const float* k2  = (const float*)d_in[2];   // Rg_RB_Bg_BR
    const float* k3  = (const float*)d_in[3];   // Rg_BR_Bg_RB
    const float* k4  = (const float*)d_in[4];   // Rb_BB_Br_RR
    float* out = (float*)d_out;

    dim3 grid(IMG_W / TX, IMG_H / TY, IMG_B);   // 16 x 64 x 16 blocks
    dim3 block(NTHREADS);
    demosaic_kernel<<<grid, block, 0, stream>>>(cfa, k1, k2, k3, k4, out);
}